// HypernymVisual_42339787604119
// MI455X (gfx1250) — compile-verified
//
#include <hip/hip_runtime.h>
#include <hip/hip_bf16.h>

typedef __attribute__((ext_vector_type(2))) float v2f;
typedef __attribute__((ext_vector_type(8))) float v8f;

#define BATCH 512
#define VFD   4096
#define HID   2048
#define EMB   300
#define NLAB  2000

// ---------------------------------------------------------------------------
// GEMM1: A2[512,2048] = relu( relu(V)[512,4096] @ W1[4096,2048] + b1 )
// One wave computes a 32(M) x 64(N) tile with V_WMMA_F32_16X16X4_F32.
// 8 waves / block, 64 blocks = 512 waves = (512/32)*(2048/64) tiles.
// ---------------------------------------------------------------------------
__global__ __launch_bounds__(256) void gemm1_relu_kernel(
    const float* __restrict__ V, const float* __restrict__ W1,
    const float* __restrict__ b1, float* __restrict__ A2)
{
    const int lane = threadIdx.x & 31;
    const int wave = blockIdx.x * (blockDim.x >> 5) + (threadIdx.x >> 5);
    const int tm = wave & 15;        // 16 M-tiles of 32
    const int tn = wave >> 4;        // 32 N-tiles of 64
    const int m0 = tm * 32;
    const int n0 = tn * 64;

    // f32 WMMA A layout (16x4): M = lane&15; lanes 0-15 hold K=0,1; 16-31 hold K=2,3
    const int mr = lane & 15;
    const int kh = (lane >> 4) << 1;   // 0 or 2

    v8f acc[2][4];
    for (int i = 0; i < 2; ++i)
        for (int j = 0; j < 4; ++j)
            for (int r = 0; r < 8; ++r) acc[i][j][r] = 0.0f;

    const float* aBase0 = V + (size_t)(m0 + mr) * VFD + kh;
    const float* aBase1 = aBase0 + (size_t)16 * VFD;

    #pragma unroll 2
    for (int k = 0; k < VFD; k += 4) {
        v2f a0, a1;
        {   const float* p = aBase0 + k;
            a0.x = fmaxf(p[0], 0.0f); a0.y = fmaxf(p[1], 0.0f); }
        {   const float* p = aBase1 + k;
            a1.x = fmaxf(p[0], 0.0f); a1.y = fmaxf(p[1], 0.0f); }

        // B layout (4x16): N = lane&15; lanes 0-15 hold K=0,1; 16-31 hold K=2,3
        v2f bf[4];
        const float* bRow = W1 + (size_t)(k + kh) * HID + (n0 + mr);
        #pragma unroll
        for (int j = 0; j < 4; ++j) {
            bf[j].x = bRow[16 * j];
            bf[j].y = bRow[16 * j + HID];
        }
        #pragma unroll
        for (int j = 0; j < 4; ++j) {
            acc[0][j] = __builtin_amdgcn_wmma_f32_16x16x4_f32(
                false, a0, false, bf[j], (short)0, acc[0][j], false, false);
            acc[1][j] = __builtin_amdgcn_wmma_f32_16x16x4_f32(
                false, a1, false, bf[j], (short)0, acc[1][j], false, false);
        }
    }

    // C/D layout: VGPR r, lanes 0-15 -> M=r, lanes 16-31 -> M=r+8; N = lane&15
    const int nc      = lane & 15;
    const int rowHalf = (lane >> 4) << 3;
    #pragma unroll
    for (int i = 0; i < 2; ++i) {
        #pragma unroll
        for (int j = 0; j < 4; ++j) {
            const int col  = n0 + 16 * j + nc;
            const float bs = b1[col];
            #pragma unroll
            for (int r = 0; r < 8; ++r) {
                const int row = m0 + 16 * i + rowHalf + r;
                A2[(size_t)row * HID + col] = fmaxf(acc[i][j][r] + bs, 0.0f);
            }
        }
    }
}

// ---------------------------------------------------------------------------
// GEMM2: E[512,300] = A2[512,2048] @ W2[2048,300] + b2
// One wave computes a 32(M) x 32(N) tile; N padded to 320 with guards.
// 8 waves / block, 20 blocks = 160 waves = 16 * 10 tiles.
// ---------------------------------------------------------------------------
__global__ __launch_bounds__(256) void gemm2_kernel(
    const float* __restrict__ A2, const float* __restrict__ W2,
    const float* __restrict__ b2, float* __restrict__ E)
{
    const int lane = threadIdx.x & 31;
    const int wave = blockIdx.x * (blockDim.x >> 5) + (threadIdx.x >> 5);
    const int tm = wave & 15;       // 16 M-tiles
    const int tn = wave >> 4;       // 10 N-tiles (covers 320 >= 300)
    const int m0 = tm * 32;
    const int n0 = tn * 32;

    const int mr = lane & 15;
    const int kh = (lane >> 4) << 1;

    v8f acc[2][2];
    for (int i = 0; i < 2; ++i)
        for (int j = 0; j < 2; ++j)
            for (int r = 0; r < 8; ++r) acc[i][j][r] = 0.0f;

    const float* aBase0 = A2 + (size_t)(m0 + mr) * HID + kh;
    const float* aBase1 = aBase0 + (size_t)16 * HID;
    const int bc0 = n0 + mr;        // B columns for this lane
    const int bc1 = n0 + 16 + mr;

    #pragma unroll 2
    for (int k = 0; k < HID; k += 4) {
        v2f a0, a1;
        {   const float* p = aBase0 + k; a0.x = p[0]; a0.y = p[1]; }
        {   const float* p = aBase1 + k; a1.x = p[0]; a1.y = p[1]; }

        const float* bRow = W2 + (size_t)(k + kh) * EMB;
        v2f bf0, bf1;
        bf0.x = (bc0 < EMB) ? bRow[bc0]       : 0.0f;
        bf0.y = (bc0 < EMB) ? bRow[bc0 + EMB] : 0.0f;
        bf1.x = (bc1 < EMB) ? bRow[bc1]       : 0.0f;
        bf1.y = (bc1 < EMB) ? bRow[bc1 + EMB] : 0.0f;

        acc[0][0] = __builtin_amdgcn_wmma_f32_16x16x4_f32(
            false, a0, false, bf0, (short)0, acc[0][0], false, false);
        acc[1][0] = __builtin_amdgcn_wmma_f32_16x16x4_f32(
            false, a1, false, bf0, (short)0, acc[1][0], false, false);
        acc[0][1] = __builtin_amdgcn_wmma_f32_16x16x4_f32(
            false, a0, false, bf1, (short)0, acc[0][1], false, false);
        acc[1][1] = __builtin_amdgcn_wmma_f32_16x16x4_f32(
            false, a1, false, bf1, (short)0, acc[1][1], false, false);
    }

    const int nc      = lane & 15;
    const int rowHalf = (lane >> 4) << 3;
    #pragma unroll
    for (int i = 0; i < 2; ++i) {
        #pragma unroll
        for (int j = 0; j < 2; ++j) {
            const int col = n0 + 16 * j + nc;
            if (col < EMB) {
                const float bs = b2[col];
                #pragma unroll
                for (int r = 0; r < 8; ++r) {
                    const int row = m0 + 16 * i + rowHalf + r;
                    E[(size_t)row * EMB + col] = acc[i][j][r] + bs;
                }
            }
        }
    }
}

// ---------------------------------------------------------------------------
// Scores: S[b,n] = -sum_d relu(E[b,d] - G[n,d])^2
// Block computes 16(b) x 32(n); E/G tiles staged transposed in LDS so the
// d-loop is broadcast + contiguous LDS reads, 3 VALU per output element.
// ---------------------------------------------------------------------------
#define TB 16
#define TN 32
__global__ __launch_bounds__(256) void score_kernel(
    const float* __restrict__ E, const float* __restrict__ G,
    float* __restrict__ S)
{
    __shared__ float Es[EMB][TB];   // 19200 B
    __shared__ float Gs[EMB][TN];   // 38400 B

    const int b0  = blockIdx.y * TB;
    const int n0  = blockIdx.x * TN;
    const int tid = threadIdx.x;

    for (int i = tid; i < TB * EMB; i += 256) {
        const int bb = i / EMB, d = i - bb * EMB;
        Es[d][bb] = E[(size_t)(b0 + bb) * EMB + d];
    }
    for (int i = tid; i < TN * EMB; i += 256) {
        const int nn = i / EMB, d = i - nn * EMB;
        const int n = n0 + nn;
        Gs[d][nn] = (n < NLAB) ? G[(size_t)n * EMB + d] : 0.0f;
    }
    __syncthreads();

    const int tx = tid & 15;        // n pair selector
    const int ty = tid >> 4;        // b within tile
    const int nn = tx * 2;

    float acc0 = 0.0f, acc1 = 0.0f;
    #pragma unroll 4
    for (int d = 0; d < EMB; ++d) {
        const float e  = Es[d][ty];
        const float g0 = Gs[d][nn];
        const float g1 = Gs[d][nn + 1];
        const float r0 = fmaxf(e - g0, 0.0f);
        const float r1 = fmaxf(e - g1, 0.0f);
        acc0 = fmaf(r0, r0, acc0);
        acc1 = fmaf(r1, r1, acc1);
    }

    const int b = b0 + ty;
    const int n = n0 + nn;
    if (n     < NLAB) S[(size_t)b * NLAB + n]     = -acc0;
    if (n + 1 < NLAB) S[(size_t)b * NLAB + n + 1] = -acc1;
}

// ---------------------------------------------------------------------------
extern "C" void kernel_launch(void* const* d_in, const int* in_sizes, int n_in,
                              void* d_out, int out_size, void* d_ws, size_t ws_size,
                              hipStream_t stream) {
    const float* vfs = (const float*)d_in[0];   // [512,4096]
    const float* W1  = (const float*)d_in[1];   // [4096,2048]
    const float* b1  = (const float*)d_in[2];   // [2048]
    const float* W2  = (const float*)d_in[3];   // [2048,300]
    const float* b2  = (const float*)d_in[4];   // [300]
    const float* gt  = (const float*)d_in[5];   // [2000,300]

    float* out   = (float*)d_out;
    float* score = out;                          // [512,2000]
    float* emb   = out + (size_t)BATCH * NLAB;   // [512,300]
    float* A2    = (float*)d_ws;                 // relu(hidden) [512,2048] = 4 MB

    // GEMM1: 512 waves (16 Mtiles x 32 Ntiles), 8 waves/block
    gemm1_relu_kernel<<<64, 256, 0, stream>>>(vfs, W1, b1, A2);
    // GEMM2: 160 waves (16 Mtiles x 10 Ntiles), 8 waves/block
    gemm2_kernel<<<20, 256, 0, stream>>>(A2, W2, b2, emb);
    // Scores: 63 x 32 blocks of 16x32 output tiles
    dim3 sg((NLAB + TN - 1) / TN, BATCH / TB);
    score_kernel<<<sg, 256, 0, stream>>>(emb, gt, score);
}